// GNNLayer_12421045420546
// MI455X (gfx1250) — compile-verified
//
#include <hip/hip_runtime.h>

typedef __attribute__((ext_vector_type(2))) float v2f;
typedef __attribute__((ext_vector_type(8))) float v8f;

#define IN_F  256
#define OUT_F 128
#define NT    8              // OUT_F / 16 N-tiles per wave
#define KP    (IN_F / 2)     // 128 k-pairs
#define SWSTR 288            // padded stride (floats) per k-pair row: 288 % 64(banks)*? -> 288 mod 64 = 32

// ---------------------------------------------------------------------------
// Kernel 1: h = tanh(features @ weight) via V_WMMA_F32_16X16X4_F32.
// Block = 256 threads (8 waves); each block computes 128 rows of h, each wave
// one 16-row M-tile across all 8 N-tiles (64 f32 accumulator VGPRs).
// Weight staged in LDS in k-pair-interleaved layout:
//   sW[kp*SWSTR + col*2 + {0,1}] = W[2kp][col], W[2kp+1][col]
// so every B fragment is ONE aligned ds_load_b64 into an even VGPR pair
// (no v_mov repacking before the WMMA), and the +SWSTR offset between the
// two lane-halves shifts banks by 32 -> conflict-free 64-bank access.
// ---------------------------------------------------------------------------
__global__ void __launch_bounds__(256) gemm_tanh_wmma(
    const float* __restrict__ A,       // [n_rows, IN_F]
    const float* __restrict__ W,       // [IN_F, OUT_F]
    const int*   __restrict__ active,  // scalar flag
    float*       __restrict__ H,       // [n_rows, OUT_F]
    int n_rows)
{
    __shared__ float sW[KP * SWSTR];   // 128*288*4 = 144 KB (<= 320 KB WGP LDS)

    const int tid = threadIdx.x;
    // Interleaving stage: pair (k even, k odd) per column, one b64 LDS store.
    for (int p = tid; p < KP * OUT_F; p += 256) {
        const int kp  = p >> 7;        // 0..127
        const int col = p & (OUT_F - 1);
        v2f w2;
        w2.x = W[(2 * kp)     * OUT_F + col];
        w2.y = W[(2 * kp + 1) * OUT_F + col];
        *(v2f*)(sW + kp * SWSTR + col * 2) = w2;
    }
    __syncthreads();

    const int wave = tid >> 5;
    const int lane = tid & 31;
    const int half = lane >> 4;        // 0: K pair {k,k+1}, 1: K pair {k+2,k+3}
    const int l    = lane & 15;
    const int Mbase = blockIdx.x * 128 + wave * 16;

    // Clamp (not branch) so EXEC stays all-ones through the WMMA loop.
    int row = Mbase + l;
    if (row >= n_rows) row = n_rows - 1;
    const float* arow  = A  + (size_t)row * IN_F + 2 * half;
    const float* wbase = sW + half * SWSTR + l * 2;

    v8f zero = {};
    v8f acc[NT];
    #pragma unroll
    for (int n = 0; n < NT; ++n) acc[n] = zero;

    for (int k = 0; k < IN_F; k += 4) {
        // A fragment 16x4 f32: lane<16 -> K=k,k+1 ; lane>=16 -> K=k+2,k+3
        v2f a = *(const v2f*)(arow + k);                    // global b64
        const float* wk = wbase + (k >> 1) * SWSTR;
        #pragma unroll
        for (int n = 0; n < NT; ++n) {
            v2f b = *(const v2f*)(wk + n * 32);             // single ds b64
            acc[n] = __builtin_amdgcn_wmma_f32_16x16x4_f32(
                false, a, false, b, (short)0, acc[n], false, false);
        }
    }

    const bool act = (active[0] != 0);
    // C/D layout: VGPR v, lanes 0-15 -> M=v, lanes 16-31 -> M=v+8, N=l
    #pragma unroll
    for (int n = 0; n < NT; ++n) {
        #pragma unroll
        for (int v = 0; v < 8; ++v) {
            int m = Mbase + v + 8 * half;
            if (m < n_rows) {
                float x = acc[n][v];
                if (act) x = tanhf(x);
                H[(size_t)m * OUT_F + n * 16 + l] = x;
            }
        }
    }
}

// ---------------------------------------------------------------------------
// Kernel 2: CSR row offsets from sorted edge_row via per-node lower_bound.
// ---------------------------------------------------------------------------
__global__ void build_rowptr(const int* __restrict__ edge_row,
                             int* __restrict__ row_ptr,
                             int n_nodes, int n_edges)
{
    int r = blockIdx.x * blockDim.x + threadIdx.x;
    if (r > n_nodes) return;
    if (r == n_nodes) { row_ptr[r] = n_edges; return; }
    int lo = 0, hi = n_edges;
    while (lo < hi) {
        int mid = (lo + hi) >> 1;
        if (edge_row[mid] < r) lo = mid + 1; else hi = mid;
    }
    row_ptr[r] = lo;
}

// ---------------------------------------------------------------------------
// Kernel 3: out[r] = sum_{e in row r} val[e] * h[col[e]].
// One wave32 per row; 32 lanes x float4 span the 128 features. No atomics.
// h (51.2 MB) is resident in the 192 MB L2, so the random gathers are L2 hits.
// ---------------------------------------------------------------------------
__global__ void __launch_bounds__(256) spmm_rowwave(
    const float* __restrict__ h,
    const int*   __restrict__ row_ptr,
    const int*   __restrict__ edge_col,
    const float* __restrict__ edge_val,
    float*       __restrict__ out,
    int n_nodes)
{
    const int wave = (int)((blockIdx.x * blockDim.x + threadIdx.x) >> 5);
    const int lane = threadIdx.x & 31;
    if (wave >= n_nodes) return;

    // Wave-uniform edge range -> per-edge col/val loads become scalar loads.
    const int e0 = __builtin_amdgcn_readfirstlane(row_ptr[wave]);
    const int e1 = __builtin_amdgcn_readfirstlane(row_ptr[wave + 1]);

    // The edge arrays are linear HBM streams: nudge them toward the caches.
    __builtin_prefetch(edge_col + e0, 0, 1);   // -> global_prefetch_b8
    __builtin_prefetch(edge_val + e0, 0, 1);

    const float* hl = h + lane * 4;
    float4 acc = {0.f, 0.f, 0.f, 0.f};
    #pragma unroll 2
    for (int e = e0; e < e1; ++e) {
        const int   col = edge_col[e];
        const float val = edge_val[e];
        const float4 hv = *(const float4*)(hl + (size_t)col * OUT_F);
        acc.x = fmaf(val, hv.x, acc.x);
        acc.y = fmaf(val, hv.y, acc.y);
        acc.z = fmaf(val, hv.z, acc.z);
        acc.w = fmaf(val, hv.w, acc.w);
    }
    *(float4*)(out + (size_t)wave * OUT_F + lane * 4) = acc;
}

// ---------------------------------------------------------------------------
extern "C" void kernel_launch(void* const* d_in, const int* in_sizes, int n_in,
                              void* d_out, int out_size, void* d_ws, size_t ws_size,
                              hipStream_t stream)
{
    (void)n_in; (void)out_size; (void)ws_size;

    const float* features = (const float*)d_in[0];
    const float* weight   = (const float*)d_in[1];
    const int*   edge_row = (const int*)d_in[2];
    const int*   edge_col = (const int*)d_in[3];
    const float* edge_val = (const float*)d_in[4];
    const int*   active   = (const int*)d_in[5];

    const int n_nodes = in_sizes[0] / IN_F;
    const int n_edges = in_sizes[2];

    // workspace layout: h [n_nodes*OUT_F f32], then row_ptr [n_nodes+1 i32]
    float* h       = (float*)d_ws;
    int*   row_ptr = (int*)((char*)d_ws + (size_t)n_nodes * OUT_F * sizeof(float));

    dim3 blk(256);

    const int gemm_blocks = (n_nodes + 127) / 128;
    gemm_tanh_wmma<<<gemm_blocks, blk, 0, stream>>>(features, weight, active, h, n_nodes);

    const int rp_blocks = (n_nodes + 1 + 255) / 256;
    build_rowptr<<<rp_blocks, blk, 0, stream>>>(edge_row, row_ptr, n_nodes, n_edges);

    const int spmm_blocks = (n_nodes + 7) / 8;   // 8 waves per block
    spmm_rowwave<<<spmm_blocks, blk, 0, stream>>>(h, row_ptr, edge_col, edge_val,
                                                  (float*)d_out, n_nodes);
}